// GAT_87385404605075
// MI455X (gfx1250) — compile-verified
//
#include <hip/hip_runtime.h>
#include <hip/hip_bf16.h>

typedef __attribute__((ext_vector_type(2))) float v2f;
typedef __attribute__((ext_vector_type(8))) float v8f;

// ---------- helpers ----------
__device__ __forceinline__ unsigned fmap(float x) {
  unsigned u = __float_as_uint(x);
  return (u & 0x80000000u) ? ~u : (u | 0x80000000u);
}
__device__ __forceinline__ float funmap(unsigned m) {
  return (m & 0x80000000u) ? __uint_as_float(m ^ 0x80000000u) : __uint_as_float(~m);
}
__device__ __forceinline__ float lrelu(float x) { return x > 0.f ? x : 0.2f * x; }

__device__ __forceinline__ void edge_ends(long long e, const long long* ei, long long E,
                                          int* s, int* d) {
  if (e < E) { *s = (int)ei[e]; *d = (int)ei[E + e]; }
  else       { int v = (int)(e - E); *s = v; *d = v; }
}

// vectorized contiguous row load: K%4==0 -> float4 (global_load_b128)
template <int K>
__device__ __forceinline__ void loadvec(const float* __restrict__ p, float* out) {
  if constexpr ((K & 3) == 0) {
#pragma unroll
    for (int i = 0; i < K / 4; ++i) {
      float4 v = reinterpret_cast<const float4*>(p)[i];
      out[4 * i + 0] = v.x; out[4 * i + 1] = v.y;
      out[4 * i + 2] = v.z; out[4 * i + 3] = v.w;
    }
  } else {
#pragma unroll
    for (int i = 0; i < K; ++i) out[i] = p[i];
  }
}

// ---------- fp32 WMMA GEMM: Out[n,fout] = A[n,fin] @ W[fin,fout] ----------
// one wave per 16-row tile; NCT column tiles of 16, A reused across them.
template <int NCT>
__global__ void gemm_f32_wmma(const float* __restrict__ A, const float* __restrict__ W,
                              float* __restrict__ Out, int n, int fin, int fout) {
  int wave = blockIdx.x * 8 + (threadIdx.x >> 5);
  int lane = threadIdx.x & 31;
  if (wave * 16 >= n) return;                 // wave-uniform: EXEC stays all-1s
  const int row  = wave * 16 + (lane & 15);   // A-matrix M per ISA layout
  const int koff = (lane >> 4) * 2;           // lanes 0-15: K0/K1, lanes 16-31: K2/K3
  const int col  = lane & 15;                 // B/D N per ISA layout

  v8f acc[NCT];
  const v8f zero = {0.f, 0.f, 0.f, 0.f, 0.f, 0.f, 0.f, 0.f};
#pragma unroll
  for (int t = 0; t < NCT; ++t) acc[t] = zero;

  for (int k0 = 0; k0 < fin; k0 += 4) {
    v2f a;
    a.x = A[(long long)row * fin + k0 + koff];
    a.y = A[(long long)row * fin + k0 + koff + 1];
#pragma unroll
    for (int t = 0; t < NCT; ++t) {
      v2f b;
      b.x = W[(long long)(k0 + koff) * fout + t * 16 + col];
      b.y = W[(long long)(k0 + koff + 1) * fout + t * 16 + col];
      acc[t] = __builtin_amdgcn_wmma_f32_16x16x4_f32(
          /*neg_a=*/false, a, /*neg_b=*/false, b,
          /*c_mod=*/(short)0, acc[t], /*reuse_a=*/false, /*reuse_b=*/false);
    }
  }
  // D layout: lane = N (col), VGPR r = row r (lanes 0-15) / r+8 (lanes 16-31)
  const int rbase = wave * 16 + ((lane >> 4) << 3);
#pragma unroll
  for (int t = 0; t < NCT; ++t)
#pragma unroll
    for (int r = 0; r < 8; ++r)
      Out[(long long)(rbase + r) * fout + t * 16 + col] = acc[t][r];
}

// ---------- per-node attention coefficients ----------
template <int H, int C>
__global__ void attn_coef(const float* __restrict__ h, const float* __restrict__ a_s,
                          const float* __restrict__ a_d, float* __restrict__ asrc,
                          float* __restrict__ adst, int n) {
  int i = blockIdx.x * blockDim.x + threadIdx.x;
  if (i >= n * H) return;
  int node = i / H, hh = i % H;
  float hv[C], as[C], ad[C];
  loadvec<C>(h + (long long)node * H * C + hh * C, hv);
  loadvec<C>(a_s + hh * C, as);
  loadvec<C>(a_d + hh * C, ad);
  float s1 = 0.f, s2 = 0.f;
#pragma unroll
  for (int c = 0; c < C; ++c) { s1 += hv[c] * as[c]; s2 += hv[c] * ad[c]; }
  asrc[i] = s1; adst[i] = s2;
}

// ---------- edge pass 1: segment max ----------
template <int H>
__global__ void edge_max_k(const long long* __restrict__ ei, long long E, long long Etot,
                           const float* __restrict__ asrc, const float* __restrict__ adst,
                           unsigned* __restrict__ emax) {
  long long e = (long long)blockIdx.x * blockDim.x + threadIdx.x;
  if (e >= Etot) return;
  int s, d; edge_ends(e, ei, E, &s, &d);
  float as[H], ad[H];
  loadvec<H>(asrc + (long long)s * H, as);
  loadvec<H>(adst + (long long)d * H, ad);
#pragma unroll
  for (int h = 0; h < H; ++h)
    atomicMax(&emax[(long long)d * H + h], fmap(lrelu(as[h] + ad[h])));
}

// ---------- edge pass 2: segment sum of exp(e - max) ----------
template <int H>
__global__ void edge_sum_k(const long long* __restrict__ ei, long long E, long long Etot,
                           const float* __restrict__ asrc, const float* __restrict__ adst,
                           const unsigned* __restrict__ emax, float* __restrict__ ssum) {
  long long e = (long long)blockIdx.x * blockDim.x + threadIdx.x;
  if (e >= Etot) return;
  int s, d; edge_ends(e, ei, E, &s, &d);
  float as[H], ad[H];
  loadvec<H>(asrc + (long long)s * H, as);
  loadvec<H>(adst + (long long)d * H, ad);
#pragma unroll
  for (int h = 0; h < H; ++h) {
    float p = __expf(lrelu(as[h] + ad[h]) - funmap(emax[(long long)d * H + h]));
    atomicAdd(&ssum[(long long)d * H + h], p);
  }
}

// ---------- edge pass 3: weighted aggregation ----------
template <int H, int C>
__global__ void edge_aggr_k(const long long* __restrict__ ei, long long E, long long Etot,
                            const float* __restrict__ asrc, const float* __restrict__ adst,
                            const unsigned* __restrict__ emax, const float* __restrict__ ssum,
                            const float* __restrict__ hlin, float* __restrict__ acc) {
  long long i = (long long)blockIdx.x * blockDim.x + threadIdx.x;
  if (i >= Etot * H) return;
  long long e = i / H; int h = (int)(i % H);
  int s, d; edge_ends(e, ei, E, &s, &d);
  float v = lrelu(asrc[(long long)s * H + h] + adst[(long long)d * H + h]);
  float p = __expf(v - funmap(emax[(long long)d * H + h]));
  float alpha = p / (ssum[(long long)d * H + h] + 1e-16f);
  float hs[C];
  loadvec<C>(hlin + (long long)s * H * C + h * C, hs);   // global_load_b128 gather
  float* ap = acc + (long long)d * H * C + h * C;
#pragma unroll
  for (int c = 0; c < C; ++c) atomicAdd(&ap[c], hs[c] * alpha);
}

// ---------- bias + optional ELU ----------
__global__ void bias_act_k(const float* __restrict__ acc, const float* __restrict__ bias,
                           float* __restrict__ out, long long total, int fout, int do_elu) {
  long long i = (long long)blockIdx.x * blockDim.x + threadIdx.x;
  if (i >= total) return;
  float v = acc[i] + bias[i % fout];
  if (do_elu) v = v > 0.f ? v : (__expf(v) - 1.f);
  out[i] = v;
}

// ---------- global mean pool (accumulate) ----------
__global__ void pool_accum_k(const float* __restrict__ hf, const long long* __restrict__ batch,
                             float* __restrict__ pooled, float* __restrict__ cnt, int n) {
  int i = blockIdx.x * blockDim.x + threadIdx.x;
  if (i >= n) return;
  int g = (int)batch[i];
  float hv[32];
  loadvec<32>(hf + (long long)i * 32, hv);
#pragma unroll
  for (int c = 0; c < 32; ++c) atomicAdd(&pooled[(long long)g * 32 + c], hv[c]);
  atomicAdd(&cnt[g], 1.f);
}

// ---------- final linear [G,C] @ [C,O] + b ----------
__global__ void final_linear_k(const float* __restrict__ pooled, const float* __restrict__ cnt,
                               const float* __restrict__ Wl, const float* __restrict__ bl,
                               float* __restrict__ out, int G, int C, int O) {
  int i = blockIdx.x * blockDim.x + threadIdx.x;
  if (i >= G * O) return;
  int g = i / O, o = i % O;
  float inv = 1.f / fmaxf(cnt[g], 1.f);
  float s = bl[o];
  for (int c = 0; c < C; ++c) s += pooled[(long long)g * C + c] * inv * Wl[c * O + o];
  out[i] = s;
}

// ---------- one GAT layer ----------
template <int Hh, int Cc>
static void run_layer(const float* hin, int fin, const float* Wm, const float* as_,
                      const float* ad_, const float* bias, float* hout, int do_elu,
                      int n, long long E, long long Etot, const long long* ei,
                      float* hlin, float* hacc, float* asrc, float* adst,
                      unsigned* emax, float* ssum, hipStream_t stream) {
  const int fout = Hh * Cc;
  const int tiles = (n + 15) / 16;
  const int gblks = (tiles + 7) / 8;                       // 8 waves / 256-thread block
  if (fout == 64)
    gemm_f32_wmma<4><<<gblks, 256, 0, stream>>>(hin, Wm, hlin, n, fin, fout);
  else
    gemm_f32_wmma<2><<<gblks, 256, 0, stream>>>(hin, Wm, hlin, n, fin, fout);

  const int nh = n * Hh;
  attn_coef<Hh, Cc><<<(nh + 255) / 256, 256, 0, stream>>>(hlin, as_, ad_, asrc, adst, n);

  hipMemsetAsync(emax, 0, (size_t)nh * sizeof(unsigned), stream);  // fmap sentinel (< any real)
  hipMemsetAsync(ssum, 0, (size_t)nh * sizeof(float), stream);
  hipMemsetAsync(hacc, 0, (size_t)n * fout * sizeof(float), stream);

  const unsigned be = (unsigned)((Etot + 255) / 256);
  edge_max_k<Hh><<<be, 256, 0, stream>>>(ei, E, Etot, asrc, adst, emax);
  edge_sum_k<Hh><<<be, 256, 0, stream>>>(ei, E, Etot, asrc, adst, emax, ssum);
  const long long teh = Etot * Hh;
  edge_aggr_k<Hh, Cc><<<(unsigned)((teh + 255) / 256), 256, 0, stream>>>(
      ei, E, Etot, asrc, adst, emax, ssum, hlin, hacc);

  const long long tot = (long long)n * fout;
  bias_act_k<<<(unsigned)((tot + 255) / 256), 256, 0, stream>>>(hacc, bias, hout, tot, fout, do_elu);
}

extern "C" void kernel_launch(void* const* d_in, const int* in_sizes, int n_in,
                              void* d_out, int out_size, void* d_ws, size_t ws_size,
                              hipStream_t stream) {
  const float*      x     = (const float*)d_in[0];
  const long long*  ei    = (const long long*)d_in[1];
  const long long*  batch = (const long long*)d_in[2];
  const float* W[3]  = {(const float*)d_in[3],  (const float*)d_in[7],  (const float*)d_in[11]};
  const float* AS[3] = {(const float*)d_in[4],  (const float*)d_in[8],  (const float*)d_in[12]};
  const float* AD[3] = {(const float*)d_in[5],  (const float*)d_in[9],  (const float*)d_in[13]};
  const float* BB[3] = {(const float*)d_in[6],  (const float*)d_in[10], (const float*)d_in[14]};
  const float* Wf  = (const float*)d_in[15];
  const float* asf = (const float*)d_in[16];
  const float* adf = (const float*)d_in[17];
  const float* bf  = (const float*)d_in[18];
  const float* Wl  = (const float*)d_in[19];
  const float* bl  = (const float*)d_in[20];

  const int FIN = 128, HC = 64, H = 8, OUTC = 32;
  const int n = in_sizes[0] / FIN;
  const long long E = in_sizes[1] / 2;
  const long long Etot = E + n;
  const int G = out_size / 2;

  // workspace carve-up (floats)
  float* p = (float*)d_ws;
  float*    hlin   = p; p += (size_t)n * HC;
  float*    hacc   = p; p += (size_t)n * HC;
  float*    hA     = p; p += (size_t)n * HC;
  float*    hB     = p; p += (size_t)n * HC;
  float*    asrc   = p; p += (size_t)n * H;
  float*    adst   = p; p += (size_t)n * H;
  unsigned* emax   = (unsigned*)p; p += (size_t)n * H;
  float*    ssum   = p; p += (size_t)n * H;
  float*    hfin   = p; p += (size_t)n * OUTC;
  float*    pooled = p; p += (size_t)G * OUTC;
  float*    cnt    = p; p += G;

  run_layer<8, 8>(x,  FIN, W[0], AS[0], AD[0], BB[0], hA, 1,
                  n, E, Etot, ei, hlin, hacc, asrc, adst, emax, ssum, stream);
  run_layer<8, 8>(hA, HC,  W[1], AS[1], AD[1], BB[1], hB, 1,
                  n, E, Etot, ei, hlin, hacc, asrc, adst, emax, ssum, stream);
  run_layer<8, 8>(hB, HC,  W[2], AS[2], AD[2], BB[2], hA, 1,
                  n, E, Etot, ei, hlin, hacc, asrc, adst, emax, ssum, stream);
  run_layer<1, 32>(hA, HC, Wf,   asf,   adf,   bf,    hfin, 0,   // 1 head, concat=False, no ELU
                   n, E, Etot, ei, hlin, hacc, asrc, adst, emax, ssum, stream);

  hipMemsetAsync(pooled, 0, ((size_t)G * OUTC + G) * sizeof(float), stream); // pooled + cnt contiguous
  pool_accum_k<<<(n + 255) / 256, 256, 0, stream>>>(hfin, batch, pooled, cnt, n);
  final_linear_k<<<(G * 2 + 255) / 256, 256, 0, stream>>>(pooled, cnt, Wl, bl, (float*)d_out, G, OUTC, 2);
}